// LocalizationLoss_8856222564605
// MI455X (gfx1250) — compile-verified
//
#include <hip/hip_runtime.h>
#include <math.h>

// ---------------------------------------------------------------------------
// LocalizationLoss on MI455X (gfx1250, wave32)
// B=1048576, N=3, C=7. Inputs: output[B,3,7] f32, target[B,3,5] f32.
// Bandwidth-bound: ~151 MB read @ 23.3 TB/s => ~6.5 us floor.
// Strategy: async b128 global->LDS staging (CDNA5 ASYNCcnt path), fused
// per-row loss, deterministic hierarchical reduction, final 512-way sum via
// V_WMMA_F32_16X16X4_F32.
// ---------------------------------------------------------------------------

#define B_TOTAL        1048576
#define NBLOCKS        512
#define NTHREADS       256
#define CHUNK          256              // rows staged per LDS pass
#define CHUNKS_PER_BLK 8                // 512 * 8 * 256 == 1048576
#define OUT_ROW        21               // floats per output row (3*7)
#define TGT_ROW        15               // floats per target row (3*5)

typedef __attribute__((ext_vector_type(2))) float v2f;
typedef __attribute__((ext_vector_type(8))) float v8f;

// Async-to-LDS builtin takes (v4i addrspace(1)*, v4i addrspace(3)*, Ii, Ii)
typedef int v4i __attribute__((vector_size(16)));
typedef __attribute__((address_space(1))) v4i gv4i;
typedef __attribute__((address_space(3))) v4i lv4i;

#if defined(__has_builtin)
#if __has_builtin(__builtin_amdgcn_global_load_async_to_lds_b128)
#define HAVE_ASYNC_B128 1
#endif
#if __has_builtin(__builtin_amdgcn_s_wait_asynccnt)
#define HAVE_WAIT_ASYNC 1
#endif
#endif

__device__ __forceinline__ void stage_b128(const float* __restrict__ g,
                                           float* __restrict__ l,
                                           int n16, int tid) {
#ifdef HAVE_ASYNC_B128
  for (int i = tid; i < n16; i += NTHREADS) {
    __builtin_amdgcn_global_load_async_to_lds_b128(
        (gv4i*)const_cast<float*>(g + i * 4),
        (lv4i*)(l + i * 4), 0, 0);
  }
#else
  const float4* gs = (const float4*)g;
  float4*       ls = (float4*)l;
  for (int i = tid; i < n16; i += NTHREADS) ls[i] = gs[i];
#endif
}

__device__ __forceinline__ void wait_stage() {
#ifdef HAVE_ASYNC_B128
#ifdef HAVE_WAIT_ASYNC
  __builtin_amdgcn_s_wait_asynccnt(0);
#else
  asm volatile("s_wait_asynccnt 0" ::: "memory");
#endif
#endif
}

__global__ void __launch_bounds__(NTHREADS)
loss_partial_kernel(const float* __restrict__ outp,
                    const float* __restrict__ tgtp,
                    float* __restrict__ partials) {
  __shared__ float so[CHUNK * OUT_ROW];   // 21504 B
  __shared__ float st[CHUNK * TGT_ROW];   // 15360 B
  __shared__ float swred[NTHREADS / 32];

  const int tid = threadIdx.x;
  float acc = 0.0f;

  const int chunk0 = blockIdx.x * CHUNKS_PER_BLK;
  for (int ci = 0; ci < CHUNKS_PER_BLK; ++ci) {
    const int chunk = chunk0 + ci;
    const float* gout = outp + (size_t)chunk * (CHUNK * OUT_ROW);
    const float* gtgt = tgtp + (size_t)chunk * (CHUNK * TGT_ROW);

    __syncthreads();   // previous pass's LDS readers are done
    stage_b128(gout, so, CHUNK * OUT_ROW / 4, tid);
    stage_b128(gtgt, st, CHUNK * TGT_ROW / 4, tid);
    wait_stage();
    __syncthreads();   // staged data visible to all waves

    const float* o = &so[tid * OUT_ROW];  // stride 21 & 15 coprime w/ 64 banks
    const float* t = &st[tid * TGT_ROW];

    float mk[3];
#pragma unroll
    for (int n = 0; n < 3; ++n) {
      const float tv = t[n * 5 + 0];
      const float p  = o[n * 7 + 0];
      // BCE term (t is exactly 0.0 or 1.0; p in (1e-4, 1-1e-4))
      const float bce = -(tv * __builtin_logf(p) +
                          (1.0f - tv) * __builtin_logf(1.0f - p));
      const float mn = (tv != 0.0f) ? 1.0f : 0.0f;
      mk[n] = mn;
      const float dx  = o[n * 7 + 1] * mn - t[n * 5 + 1];
      const float dy  = o[n * 7 + 2] * mn - t[n * 5 + 2];
      // where(mask, sqrt(out_wh), 0): out_wh > 0 so mn*sqrt is exact
      const float sw  = mn * __builtin_sqrtf(o[n * 7 + 3]);
      const float dwh = sw - __builtin_sqrtf(t[n * 5 + 3]);
      // weights folded: 5*(Lx+Ly) + 10*Lwh + 0.5*bce
      acc += 5.0f * (dx * dx + dy * dy) + 10.0f * (dwh * dwh) + 0.5f * bce;
    }

    // CE: log_softmax over the N axis of masked logits, 3 columns
#pragma unroll
    for (int c = 0; c < 3; ++c) {
      const float x0 = o[0 * 7 + 4 + c] * mk[0];
      const float x1 = o[1 * 7 + 4 + c] * mk[1];
      const float x2 = o[2 * 7 + 4 + c] * mk[2];
      const float mx = fmaxf(x0, fmaxf(x1, x2));
      const float lse = mx + __builtin_logf(__builtin_expf(x0 - mx) +
                                            __builtin_expf(x1 - mx) +
                                            __builtin_expf(x2 - mx));
      const int idx = (int)t[c * 5 + 4];   // cls in {0,1,2}
      const float xi = (idx == 0) ? x0 : ((idx == 1) ? x1 : x2);
      acc += 3.0f * (lse - xi);
    }
  }

  // Deterministic block reduction: wave32 shfl tree, then first wave over LDS.
#pragma unroll
  for (int off = 16; off > 0; off >>= 1) acc += __shfl_xor(acc, off, 32);
  const int wave = tid >> 5, lane = tid & 31;
  if (lane == 0) swred[wave] = acc;
  __syncthreads();
  if (tid < NTHREADS / 32) {
    float v = swred[tid];
#pragma unroll
    for (int off = (NTHREADS / 64); off > 0; off >>= 1)
      v += __shfl_xor(v, off, NTHREADS / 32);
    if (tid == 0) partials[blockIdx.x] = v;
  }
}

// Final reduction: one wave32. 8 chained V_WMMA_F32_16X16X4_F32 with B = ones
// sum all 512 block partials into the f32 C tile (fixed order => deterministic).
__global__ void __launch_bounds__(32)
loss_final_kernel(const float* __restrict__ partials, float* __restrict__ out) {
  const int l = threadIdx.x;             // 0..31, EXEC all ones for WMMA
  v8f c = {0.f, 0.f, 0.f, 0.f, 0.f, 0.f, 0.f, 0.f};
  v2f ones; ones[0] = 1.0f; ones[1] = 1.0f;

#pragma unroll
  for (int g = 0; g < NBLOCKS / 64; ++g) {
    v2f a;
    a[0] = partials[g * 64 + l];
    a[1] = partials[g * 64 + 32 + l];
    // D = A(16x4) * ones(4x16) + C : every column of D gets per-row sums
    c = __builtin_amdgcn_wmma_f32_16x16x4_f32(false, a, false, ones,
                                              (short)0, c, false, false);
  }
  // Column n of C is identical for all n. Lanes 0-15 hold rows 0-7 in c[0..7],
  // lanes 16-31 hold rows 8-15. Sum components, then add across the half-waves.
  float s = c[0] + c[1] + c[2] + c[3] + c[4] + c[5] + c[6] + c[7];
  s += __shfl_xor(s, 16, 32);
  if (l == 0)
    out[0] = s * (1.0f / (3.0f * (float)B_TOTAL)) + 0.5f;
}

extern "C" void kernel_launch(void* const* d_in, const int* in_sizes, int n_in,
                              void* d_out, int out_size, void* d_ws, size_t ws_size,
                              hipStream_t stream) {
  const float* outp = (const float*)d_in[0];   // output [B,3,7] f32
  const float* tgtp = (const float*)d_in[1];   // target [B,3,5] f32
  float* partials = (float*)d_ws;              // 512 floats of scratch
  loss_partial_kernel<<<NBLOCKS, NTHREADS, 0, stream>>>(outp, tgtp, partials);
  loss_final_kernel<<<1, 32, 0, stream>>>(partials, (float*)d_out);
}